// HDRRenderer_7842610282847
// MI455X (gfx1250) — compile-verified
//
#include <hip/hip_runtime.h>
#include <cmath>

// ---------------------------------------------------------------------------
// HDR env-lit PBR renderer for MI455X (gfx1250, wave32).
//
// D = A x B via V_WMMA_F32_16X16X4_F32:
//   NL[16x16] = normals(16x3, K padded to 4) x ldirs(3x16)  -> n.l
//   VL[16x16] = viewdirs(16x3)               x ldirs(3x16)  -> v.l
// With unit n,v,l:  |v+l| = sqrt(2+2 v.l),  NdotH=(n.v+n.l)/|v+l|,
// VdotH=(1+v.l)/|v+l| -> the Cook-Torrance inner loop needs no 3-vectors.
// All per-row-invariant algebra hoisted; divides/sqrt use v_rcp/v_rsq.
// ---------------------------------------------------------------------------

typedef __attribute__((ext_vector_type(2))) float v2f;
typedef __attribute__((ext_vector_type(8))) float v8f;

#define EPSF 1e-6f
#define PIF  3.14159265358979323846f

#define ROWS_PER_TILE   16
#define WAVES_PER_BLOCK 4
#define BLOCK_THREADS   (WAVES_PER_BLOCK * 32)
#define M_MAX           512

__global__ __launch_bounds__(BLOCK_THREADS)
void hdr_render_kernel(const float* __restrict__ normals,     // (N,3)
                       const float* __restrict__ albedo,      // (N,3)
                       const float* __restrict__ metallic,    // (N,1)
                       const float* __restrict__ smoothness,  // (N,1)
                       const float* __restrict__ viewdirs,    // (NV,3)
                       const float* __restrict__ envvis,      // (N,M)
                       const float* __restrict__ envmap,      // (W,H,3)
                       float* __restrict__ out,               // (NV,3)
                       int N, int V, int M, int W, int H, int NV, int numTiles)
{
    // Per-block LDS tables: light dirs, solid angle, radiance (exact-pixel lookup).
    __shared__ float s_lx[M_MAX], s_ly[M_MAX], s_lz[M_MAX], s_sa[M_MAX];
    __shared__ float s_Lr[M_MAX], s_Lg[M_MAX], s_Lb[M_MAX];
    // Per-row (per (n,v) pair) prepared data, one tile per wave.
    __shared__ float s_row[WAVES_PER_BLOCK][ROWS_PER_TILE][16];

    const int tid = threadIdx.x;

    // ---- cooperative env-sample table build (once per block) ----
    for (int m = tid; m < M; m += BLOCK_THREADS) {
        int i = m % W, j = m / W;                    // m = j*W + i
        float u  = (i + 0.5f) / (float)W;
        float vq = (j + 0.5f) / (float)H;
        float phi   = u * 2.0f * PIF + 0.5f * PIF;
        float theta = vq * PIF;
        float st = sinf(theta), ct = cosf(theta);
        float sp = sinf(phi),   cp = cosf(phi);
        s_lx[m] = st * cp;
        s_ly[m] = ct;
        s_lz[m] = -st * sp;
        s_sa[m] = st * (PIF / (float)H) * (2.0f * PIF / (float)W);
        // bilinear grid_sample degenerates to exact pixel (i,j)
        const float* em = envmap + ((size_t)i * H + j) * 3;
        s_Lr[m] = em[0]; s_Lg[m] = em[1]; s_Lb[m] = em[2];
    }

    const int wave = tid >> 5;
    const int lane = tid & 31;
    const int tile = blockIdx.x * WAVES_PER_BLOCK + wave;

    // ---- per-row setup: lane j (<16) prepares row j of this wave's tile ----
    if (tile < numTiles && lane < ROWS_PER_TILE) {
        int grow = tile * ROWS_PER_TILE + lane;          // (n,v) flat row
        int gcl  = grow < NV ? grow : NV - 1;            // clamp for ragged tail
        int nIdx = gcl / V;
        float nx = normals[nIdx*3+0], ny = normals[nIdx*3+1], nz = normals[nIdx*3+2];
        float ni = 1.0f / fmaxf(sqrtf(nx*nx + ny*ny + nz*nz), 1e-12f);
        nx *= ni; ny *= ni; nz *= ni;
        float vx = viewdirs[gcl*3+0], vy = viewdirs[gcl*3+1], vz = viewdirs[gcl*3+2];
        float vi = 1.0f / fmaxf(sqrtf(vx*vx + vy*vy + vz*vz), 1e-12f);
        vx *= vi; vy *= vi; vz *= vi;
        float met   = metallic[nIdx];
        float rough = 1.0f - smoothness[nIdx];
        float alpha = rough * rough;
        float ar = albedo[nIdx*3+0], ag = albedo[nIdx*3+1], ab = albedo[nIdx*3+2];
        float* rd = s_row[wave][lane];
        rd[0]  = nx*vx + ny*vy + nz*vz;                  // raw n.v
        rd[1]  = alpha;
        rd[2]  = 0.04f + (ar - 0.04f) * met;             // f0.rgb
        rd[3]  = 0.04f + (ag - 0.04f) * met;
        rd[4]  = 0.04f + (ab - 0.04f) * met;
        float dm = (1.0f - met) * (1.0f / PIF);          // diffuse base
        rd[5]  = ar * dm; rd[6] = ag * dm; rd[7] = ab * dm;
        rd[8]  = nx; rd[9]  = ny; rd[10] = nz;
        rd[11] = vx; rd[12] = vy; rd[13] = vz;
        rd[14] = (float)nIdx;
        rd[15] = 0.0f;
    }
    __syncthreads();

    if (tile >= numTiles) return;   // wave-uniform exit (EXEC stays all-ones)

    const int half = lane >> 4;     // A/B K-group: lanes 0-15 -> K0,K1; 16-31 -> K2,K3
    const int l15  = lane & 15;

    // A matrices: row = l15, components (x,y,z,0) along K.
    // ISA 16x4 f32 A layout: VGPR0 = K0 (half0) / K2 (half1); VGPR1 = K1 / K3.
    const float* myrow = s_row[wave][l15];
    v2f An, Av;
    if (half == 0) { An.x = myrow[8];  An.y = myrow[9];
                     Av.x = myrow[11]; Av.y = myrow[12]; }
    else           { An.x = myrow[10]; An.y = 0.0f;
                     Av.x = myrow[13]; Av.y = 0.0f; }

    // Hoist row-invariant BRDF algebra for the 8 rows this lane owns in D
    // (D-matrix row = r + 8*half, column = l15).
    float ndv[8];                      // raw n.v
    float a2[8], a2m1[8];              // alpha^2, alpha^2-1
    float onemk[8], kEps[8];           // 1-k, k+EPS        (k = alpha/2)
    float Gv[8];                       // NdotV-half of the geometry term
    float c4v[8];                      // 4*NdotV
    float f0r[8], f0g[8], f0b[8];      // Fresnel F0
    float dbr[8], dbg[8], dbb[8];      // diffuse base (1-met)*albedo/pi
    int   voff[8];                     // env_vis row offsets
#pragma unroll
    for (int r = 0; r < 8; ++r) {
        const float* rd = s_row[wave][r + 8*half];
        ndv[r] = rd[0];
        float a = rd[1];
        a2[r]   = a * a;
        a2m1[r] = a2[r] - 1.0f;
        float k  = 0.5f * a;
        onemk[r] = 1.0f - k;
        kEps[r]  = k + EPSF;
        float NdotV = fmaxf(rd[0], EPSF);
        Gv[r]  = NdotV / (NdotV * onemk[r] + kEps[r]);   // once per row: precise
        c4v[r] = 4.0f * NdotV;
        f0r[r] = rd[2];  f0g[r] = rd[3];  f0b[r] = rd[4];
        dbr[r] = rd[5];  dbg[r] = rd[6];  dbb[r] = rd[7];
        voff[r] = (int)rd[14] * M;
    }

    float accr[8] = {0,0,0,0,0,0,0,0};
    float accg[8] = {0,0,0,0,0,0,0,0};
    float accb[8] = {0,0,0,0,0,0,0,0};

    const int mtiles = M >> 4;
    for (int mt = 0; mt < mtiles; ++mt) {
        const int m = (mt << 4) + l15;                 // this lane's sample column

        // B matrix: col = l15, K rows = light-dir comps (x,y,z,0).
        v2f Bl;
        if (half == 0) { Bl.x = s_lx[m]; Bl.y = s_ly[m]; }
        else           { Bl.x = s_lz[m]; Bl.y = 0.0f;  }

        v8f cz = {};
        // 8-arg form: (neg_a, A, neg_b, B, c_mod, C, reuse_a, reuse_b)
        v8f NL = __builtin_amdgcn_wmma_f32_16x16x4_f32(false, An, false, Bl,
                                                       (short)0, cz, false, false);
        v8f VL = __builtin_amdgcn_wmma_f32_16x16x4_f32(false, Av, false, Bl,
                                                       (short)0, cz, false, false);

        const float sa = s_sa[m];
        const float Lr = s_Lr[m], Lg = s_Lg[m], Lb = s_Lb[m];

#pragma unroll
        for (int r = 0; r < 8; ++r) {
            float nl_raw = NL[r];
            float vl     = VL[r];
            float NdotL  = fmaxf(nl_raw, EPSF);
            // 1/|v+l| = rsq(2 + 2 v.l); clamp matches ref's max(|v+l|,1e-12)
            float rs     = __builtin_amdgcn_rsqf(fmaxf(fmaf(2.0f, vl, 2.0f), 1e-24f));
            float NdotH  = fmaxf((ndv[r] + nl_raw) * rs, 0.0f);
            float VdotH  = fmaxf((1.0f + vl) * rs, 0.0f);
            float t  = 1.0f - fminf(VdotH, 1.0f);
            float t2 = t * t;
            float p5 = t2 * t2 * t;                    // (1-VdotH)^5
            float q  = 1.0f - p5;
            float den = fmaf(NdotH * NdotH, a2m1[r], 1.0f);
            float Dd  = a2[r] * __builtin_amdgcn_rcpf(fmaf(PIF * den, den, EPSF));
            float Gl  = NdotL * __builtin_amdgcn_rcpf(fmaf(NdotL, onemk[r], kEps[r]));
            float specS = Dd * Gv[r] * Gl *
                          __builtin_amdgcn_rcpf(fmaf(c4v[r], NdotL, EPSF));
            float w   = envvis[voff[r] + m] * NdotL * sa;   // coalesced over lanes
            float Lrw = Lr * w, Lgw = Lg * w, Lbw = Lb * w;
            // Fr = f0 + (1-f0)*p5 = fma(f0, 1-p5, p5)
            float Frr = fmaf(f0r[r], q, p5);
            float Frg = fmaf(f0g[r], q, p5);
            float Frb = fmaf(f0b[r], q, p5);
            // (1-Fr)*db + Fr*spec = fma(Fr, spec-db, db)
            accr[r] = fmaf(fmaf(Frr, specS - dbr[r], dbr[r]), Lrw, accr[r]);
            accg[r] = fmaf(fmaf(Frg, specS - dbg[r], dbg[r]), Lgw, accg[r]);
            accb[r] = fmaf(fmaf(Frb, specS - dbb[r], dbb[r]), Lbw, accb[r]);
        }
    }

    // Column reduction: rows 0-7 live in lanes 0-15, rows 8-15 in lanes 16-31;
    // xor masks <16 keep the two halves independent.
#pragma unroll
    for (int r = 0; r < 8; ++r) {
        for (int s = 1; s < 16; s <<= 1) {
            accr[r] += __shfl_xor(accr[r], s, 32);
            accg[r] += __shfl_xor(accg[r], s, 32);
            accb[r] += __shfl_xor(accb[r], s, 32);
        }
    }

    if (l15 == 0) {
#pragma unroll
        for (int r = 0; r < 8; ++r) {
            int grow = tile * ROWS_PER_TILE + r + 8*half;
            if (grow < NV) {
                out[grow*3 + 0] = accr[r];
                out[grow*3 + 1] = accg[r];
                out[grow*3 + 2] = accb[r];
            }
        }
    }
}

extern "C" void kernel_launch(void* const* d_in, const int* in_sizes, int n_in,
                              void* d_out, int out_size, void* d_ws, size_t ws_size,
                              hipStream_t stream) {
    const float* normals  = (const float*)d_in[0];
    const float* albedo   = (const float*)d_in[1];
    const float* metallic = (const float*)d_in[2];
    const float* smooth   = (const float*)d_in[3];
    const float* viewd    = (const float*)d_in[4];
    const float* envvis   = (const float*)d_in[5];
    const float* envmap   = (const float*)d_in[6];
    float* out = (float*)d_out;

    const int N  = in_sizes[0] / 3;          // B == 1
    const int NV = in_sizes[4] / 3;          // N*V rows
    const int V  = NV / N;
    const int M  = in_sizes[5] / N;          // env samples (512)
    // setup uses W = 2*H, W*H = M  ->  H = sqrt(M/2)
    int H = (int)(sqrtf((float)M * 0.5f) + 0.5f);
    if (H < 1) H = 1;
    const int W = M / H;

    const int numTiles = (NV + ROWS_PER_TILE - 1) / ROWS_PER_TILE;
    const int blocks   = (numTiles + WAVES_PER_BLOCK - 1) / WAVES_PER_BLOCK;

    hdr_render_kernel<<<blocks, BLOCK_THREADS, 0, stream>>>(
        normals, albedo, metallic, smooth, viewd, envvis, envmap, out,
        N, V, M, W, H, NV, numTiles);
}